// GraphConv_82815559402084
// MI455X (gfx1250) — compile-verified
//
#include <hip/hip_runtime.h>

#define B_ 16
#define N_ 4096
#define DIN_ 128
#define DOUT_ 128
#define C_ 3
#define E_ 65536

typedef float v2f __attribute__((ext_vector_type(2)));
typedef float v8f __attribute__((ext_vector_type(8)));

// ---------------------------------------------------------------------------
// Kernel 0: transpose W[c,d,o] -> WT[c,o,d] so WMMA B-fragments are contiguous
// ---------------------------------------------------------------------------
__global__ __launch_bounds__(256) void wt_kernel(const float* __restrict__ W,
                                                 float* __restrict__ WT) {
    int idx = blockIdx.x * 256 + threadIdx.x;      // [0, C*128*128)
    int c = idx >> 14;
    int d = (idx >> 7) & 127;
    int o = idx & 127;
    WT[(c << 14) + (o << 7) + d] = W[idx];
}

// ---------------------------------------------------------------------------
// Kernel 1: fw[b,c] = x[b] @ W[c] + bias[c]   via V_WMMA_F32_16X16X4_F32
// Block = 256 threads (8 waves). Wave w computes rows [m0, m0+16) x all 128
// output cols as 8 chained 16x16 tiles, K accumulated 4 at a time.
// ---------------------------------------------------------------------------
__global__ __launch_bounds__(256) void gemm_fw_kernel(const float* __restrict__ x,
                                                      const float* __restrict__ WT,
                                                      const float* __restrict__ bias,
                                                      float* __restrict__ fw) {
    const int lane = threadIdx.x & 31;
    const int wave = threadIdx.x >> 5;
    const int bc   = blockIdx.y;          // b*C + c
    const int b    = bc / C_;
    const int c    = bc - b * C_;
    const int m0   = (blockIdx.x * 8 + wave) * 16;

    const int half = lane >> 4;           // 0: K=k0..k0+1, 1: K=k0+2..k0+3
    const int l15  = lane & 15;

    const float* aptr = x  + ((size_t)b * N_ + m0 + l15) * DIN_ + half * 2;
    const float* wtc  = WT + ((size_t)c << 14);     // WT[c, o, d], 128x128

    v8f acc[8];
#pragma unroll
    for (int t = 0; t < 8; ++t) acc[t] = (v8f)(0.0f);

    for (int k0 = 0; k0 < DIN_; k0 += 4) {
        v2f a = *(const v2f*)(aptr + k0);           // A frag: x rows m0..m0+15
#pragma unroll
        for (int t = 0; t < 8; ++t) {
            // B frag: W[k0+dk, n0+l15] = WT[n0+l15, k0+dk] -> contiguous pair
            v2f bb = *(const v2f*)(wtc + (size_t)(t * 16 + l15) * DIN_ + k0 + half * 2);
            acc[t] = __builtin_amdgcn_wmma_f32_16x16x4_f32(
                false, a, false, bb, (short)0, acc[t], false, false);
        }
    }

    // D layout: VGPR r -> row (r + half*8), lane l15 -> col (t*16 + l15)
    float* outp = fw + ((size_t)bc * N_ + m0) * DOUT_;
#pragma unroll
    for (int t = 0; t < 8; ++t) {
        const int col = t * 16 + l15;
        const float bv = bias[c * DOUT_ + col];
#pragma unroll
        for (int r = 0; r < 8; ++r) {
            const int row = r + half * 8;
            outp[(size_t)row * DOUT_ + col] = acc[t][r] + bv;
        }
    }
}

// ---------------------------------------------------------------------------
// Kernel 2: edge aggregation. One wave per edge: lane t handles output
// elements [4t, 4t+4): float4 gather from fw (L2-resident), 4 atomic f32 adds.
// out[b,dst] += val * fw[b,c,src]; the sum over c happens via the atomics.
// ---------------------------------------------------------------------------
__global__ __launch_bounds__(256) void aggregate_kernel(const int* __restrict__ eidx,
                                                        const float* __restrict__ eval_,
                                                        const float* __restrict__ fw,
                                                        float* __restrict__ out) {
    const long long edge = (long long)blockIdx.x * 8 + (threadIdx.x >> 5);
    const int lane = threadIdx.x & 31;

    const long long bc = edge >> 16;            // E = 65536
    const int b = (int)(bc / C_);

    const int2 ij = ((const int2*)eidx)[edge];  // .x = dst row i, .y = src col j
    const float v = eval_[edge];

    const float4 s = *(const float4*)(fw + (bc << 19) + ((size_t)ij.y << 7) + lane * 4);

    float* o = out + ((size_t)b << 19) + ((size_t)ij.x << 7) + lane * 4;
    unsafeAtomicAdd(o + 0, s.x * v);
    unsafeAtomicAdd(o + 1, s.y * v);
    unsafeAtomicAdd(o + 2, s.z * v);
    unsafeAtomicAdd(o + 3, s.w * v);
}

// ---------------------------------------------------------------------------
extern "C" void kernel_launch(void* const* d_in, const int* in_sizes, int n_in,
                              void* d_out, int out_size, void* d_ws, size_t ws_size,
                              hipStream_t stream) {
    const float* x    = (const float*)d_in[0];   // [B,N,DIN]
    const float* W    = (const float*)d_in[1];   // [C,DIN,DOUT]
    const float* bias = (const float*)d_in[2];   // [C,DOUT]
    const int*   eidx = (const int*)d_in[3];     // [B,C,E,2]
    const float* ev   = (const float*)d_in[4];   // [B,C,E]
    float* out = (float*)d_out;                  // [B,N,DOUT]

    // ws layout: [ WT : C*128*128 f32 = 196608 B ][ fw : B*C*N*128 f32 = 100.66 MB ]
    float* WT = (float*)d_ws;
    float* fw = (float*)((char*)d_ws + (size_t)C_ * DIN_ * DOUT_ * sizeof(float));

    // out is poisoned by the harness; zero it every call (graph-capture safe).
    hipMemsetAsync(d_out, 0, (size_t)out_size * sizeof(float), stream);

    // Transpose W (196 KB)
    wt_kernel<<<(C_ * DIN_ * DOUT_) / 256, 256, 0, stream>>>(W, WT);

    // 48 GEMMs: grid = (N / 128 rows-per-block, B*C)
    dim3 ggrid(N_ / 128, B_ * C_);
    gemm_fw_kernel<<<ggrid, 256, 0, stream>>>(x, WT, bias, fw);

    // 3.1M edges, 8 edges (waves) per 256-thread block
    const long long edges = (long long)B_ * C_ * E_;
    aggregate_kernel<<<(unsigned)(edges / 8), 256, 0, stream>>>(eidx, ev, fw, out);
}